// CritiGraph_66391604462097
// MI455X (gfx1250) — compile-verified
//
#include <hip/hip_runtime.h>
#include <hip/hip_bf16.h>

// ---------------------------------------------------------------------------
// CritiGraph for MI455X (gfx1250, wave32).
//
// 256 workgroups (one per (b,t1)) x 256 threads (8 wave32 waves).
// Phase A1: ALL global traffic staged into LDS with per-lane
//           global_load_async_to_lds_b128/b32 (CDNA5 ASYNCcnt path), one
//           s_wait_asynccnt + barrier.
// Phase A2: (a) all 512 (t2,tp) setup elements in parallel, tp-group sum via
//           3x __shfl_xor (wave32 lane permute); packs {abs(pos), w*p, w*y}
//           into a float4 per element. (b) 128x8 candidate xor-masks hashed
//           once into a u16 LDS table (shared by cand and its negated twin,
//           and by Phase C). (c) s_oa[tp] = |sta[perm[tp]]|.
// Phase B:  per (cand,tp) pair: candidate = 1 ds_load_u16 + 2 xors; 64-long
//           reduction at 1x ds_load_b128 + ~8 VALU per step, dual
//           accumulators to halve the FMA chain.
// Phase C:  packed-u64 LDS atomicMin argmin (first-min tie-break).
// ---------------------------------------------------------------------------

namespace {
constexpr int H   = 16;
constexpr int TP  = 8;
constexpr int K   = 8;          // C*H/DIV
constexpr int HK  = H * K;      // 128
constexpr int Bsz = 4;
constexpr int T1  = 64;
constexpr int T2  = 64;
constexpr int NCAND = 2 * HK + 1;  // 257
constexpr int BLOCK = 256;

__device__ __forceinline__ unsigned mix32(unsigned x) {
  x ^= x >> 16; x *= 0x7feb352dU;
  x ^= x >> 15; x *= 0x846ca68bU;
  x ^= x >> 16; return x;
}
__device__ __forceinline__ unsigned rng4(unsigned a, unsigned b, unsigned c, unsigned d) {
  return mix32(a * 0x9E3779B9u + mix32(b * 0x85EBCA6Bu + mix32(c * 0xC2B2AE35u + mix32(d ^ 0x27220A95u))));
}
} // namespace

__global__ void __launch_bounds__(BLOCK)
critigraph_kernel(const int* __restrict__ sta, const int* __restrict__ pos,
                  const float* __restrict__ val_v, const float* __restrict__ val_n,
                  const int* __restrict__ locations, const int* __restrict__ epoch_p,
                  float* __restrict__ out) {
  __shared__ int            s_pg[T2][TP];   // raw gathered pos locations
  __shared__ float4         s_q[T2][TP];    // {bits(abs(pos)), w*p, w*y, 0}
  __shared__ unsigned short s_msk[HK][TP];  // per-(flip,tree-column) xor masks
  __shared__ float          s_vv[T2];
  __shared__ float          s_vn[T2];
  __shared__ int            s_sta[TP];
  __shared__ int            s_perm[TP];
  __shared__ int            s_oa[TP];       // |sta[perm[tp]]|
  __shared__ unsigned long long s_best[TP];

  const unsigned row = blockIdx.x;    // b*T1 + t1
  const int tid = threadIdx.x;
  const int b   = (int)(row >> 6);    // T1 == 64

  // ---- Phase A0: tiny per-row setup --------------------------------------
  if (tid < TP) {
    s_best[tid] = ~0ull;
    s_sta[tid]  = locations[(size_t)sta[row] * TP + tid];
  }
  if (tid == 0) {                     // global tree permutation (Fisher-Yates)
    int p8[TP];
    for (int i = 0; i < TP; ++i) p8[i] = i;
    for (int i = TP - 1; i > 0; --i) {
      const int j = (int)(rng4(0x5045524Du, (unsigned)i, 42u, 7u) % (unsigned)(i + 1));
      const int t = p8[i]; p8[i] = p8[j]; p8[j] = t;
    }
    for (int i = 0; i < TP; ++i) s_perm[i] = p8[i];
  }

  // ---- Phase A1: async stage ALL bulk inputs into LDS --------------------
  if (tid < T2 * 2) {
    // per-lane scattered gather locations[pos[b,t2]][:] -> LDS (ASYNCcnt)
    const int t2   = tid >> 1;
    const int half = tid & 1;
    const int pidx = pos[b * T2 + t2];
    const int* g   = locations + (size_t)pidx * TP + half * 4;
#if defined(__gfx1250__)
    const unsigned lds = (unsigned)(size_t)(const void*)&s_pg[t2][half * 4];
    asm volatile("global_load_async_to_lds_b128 %0, %1, off"
                 :: "v"(lds), "v"(g) : "memory");
#else
    *(int4*)&s_pg[t2][half * 4] = *(const int4*)g;
#endif
  } else if (tid < 192) {
    const int t2 = tid - 128;
    const float* g = val_v + row * T2 + t2;
#if defined(__gfx1250__)
    const unsigned lds = (unsigned)(size_t)(const void*)&s_vv[t2];
    asm volatile("global_load_async_to_lds_b32 %0, %1, off"
                 :: "v"(lds), "v"(g) : "memory");
#else
    s_vv[t2] = *g;
#endif
  } else {
    const int t2 = tid - 192;
    const float* g = val_n + row * T2 + t2;
#if defined(__gfx1250__)
    const unsigned lds = (unsigned)(size_t)(const void*)&s_vn[t2];
    asm volatile("global_load_async_to_lds_b32 %0, %1, off"
                 :: "v"(lds), "v"(g) : "memory");
#else
    s_vn[t2] = *g;
#endif
  }
#if defined(__gfx1250__)
  asm volatile("s_wait_asynccnt 0" ::: "memory");
#endif
  __syncthreads();

  // ---- Phase A2a: per-(t2,tp) derived quantities, all threads ------------
  const int ep = epoch_p[0];
  // neighbor mask (epoch <= 5): keep full row w.p. 0.8 else one column
  const unsigned hm = rng4(row, 0x6D6173u, 0x6Bu, 99u);
  const int  chosen = hm >= 858993460u;                        // ~ p>0.2
  const int  t2sel  = (int)(rng4(row, 0x743273u, 0x656Cu, 3u) & (T2 - 1));
  const int  full   = ep > 5;

  for (int el = tid; el < T2 * TP; el += BLOCK) {
    const int t2 = el >> 3;
    const int tp = el & 7;
    const int c2 = s_pg[t2][tp];
    const int c1 = s_sta[tp];
    const float vn = s_vn[t2];
    const float vv = s_vv[t2];
    const int a1 = c1 < 0 ? -c1 : c1;
    const int a2 = c2 < 0 ? -c2 : c2;
    const int xr = a1 ^ a2;
    // s(x) = (32 - clz(x+1))/16  ->  (1-s) = (clz(x+1)-16)/16
    const float f  = (float)(__clz(xr + 1) - 16) * 0.0625f;
    const float sg = ((c1 ^ c2) < 0) ? -1.f : 1.f;             // sign(0)=+1
    const float d  = sg * f * vn;                              // d_sp[t2][tp]
    // sum over the 8-lane tp group (lane&7 == tp by construction)
    float dsum = d;
#pragma unroll
    for (int m = 1; m < 8; m <<= 1) dsum += __shfl_xor(dsum, m, 8);
    const float w = (full || chosen || t2 == t2sel) ? 1.f : 0.f;
    // w in {0,1}: fold into both terms, so loss term = (w*p + w*q)^2
    float4 q;
    q.x = __int_as_float(a2);
    q.y = w * ((dsum - d) * 0.125f - vv);                      // w * p
    q.z = w * ((c2 < 0) ? -vn : vn) * 0.0078125f;              // w*sgn(pos)*vn/(TP*16)
    q.w = 0.f;
    s_q[t2][tp] = q;
  }

  // ---- Phase A2b: hash all candidate xor-masks once (mask < 2^i <= 2^15) -
  for (int el = tid; el < HK * TP; el += BLOCK) {
    const int j  = el >> 3;          // flip index: i*K + k
    const int tp = el & 7;
    const int i  = j >> 3;           // K == 8
    const int k  = j & 7;
    const int src = s_perm[tp];
    const unsigned m = rng4(row + 0x1000u, (unsigned)(i * K + k), (unsigned)src, 0xA5A5A5u)
                       & ((1u << i) - 1u);
    s_msk[j][tp] = (unsigned short)m;
  }
  // ---- Phase A2c: permuted-column |sta| ----------------------------------
  if (tid < TP) {
    const int sv = s_sta[s_perm[tid]];
    s_oa[tid] = sv < 0 ? -sv : sv;
  }
  __syncthreads();

  // ---- Phase B: score every (cand, tp) pair ------------------------------
  for (int pair = tid; pair < NCAND * TP; pair += BLOCK) {
    const int tp   = pair & 7;
    const int cand = pair >> 3;
    const int oa   = s_oa[tp];
    int c1;
    if (cand == HK) {
      c1 = oa;                                  // middle candidate: |sta|
    } else {
      int j = cand, neg = 0;
      if (j > HK) { j -= HK + 1; neg = 1; }
      const int i = j >> 3;
      const int r = (oa ^ (1 << i)) ^ (int)s_msk[j][tp];
      c1 = neg ? -r : r;
    }
    const int c1a  = c1 < 0 ? -c1 : c1;
    const int smsk = c1 < 0 ? -1 : 0;           // candidate-sign xor/sub mask
    float l0 = 0.f, l1 = 0.f;                   // dual accumulators (chain /2)
#pragma unroll 8
    for (int t2 = 0; t2 < T2; t2 += 2) {
      const float4 qa = s_q[t2][tp];            // ds_load_b128
      const float4 qb = s_q[t2 + 1][tp];        // ds_load_b128
      int na = __clz((c1a ^ __float_as_int(qa.x)) + 1) - 16;
      int nb = __clz((c1a ^ __float_as_int(qb.x)) + 1) - 16;
      na = (na ^ smsk) - smsk;
      nb = (nb ^ smsk) - smsk;
      const float ta = fmaf((float)na, qa.z, qa.y);   // w*(p+q)
      const float tb = fmaf((float)nb, qb.z, qb.y);
      l0 = fmaf(ta, ta, l0);
      l1 = fmaf(tb, tb, l1);
    }
    const float loss = l0 + l1;
    // loss >= 0: IEEE bits order-preserving; low word = cand gives
    // first-index tie-break, matching jnp.argmin.
    const unsigned long long key =
        ((unsigned long long)__float_as_uint(loss) << 32) | (unsigned)cand;
    atomicMin(&s_best[tp], key);
  }
  __syncthreads();

  // ---- Phase C: emit winner per tp ---------------------------------------
  if (tid < TP) {
    const int tp = tid;
    const unsigned long long bk = s_best[tp];
    const int   cand  = (int)(bk & 0xffffffffu);
    const float mloss = __uint_as_float((unsigned)(bk >> 32));
    const int   oa    = s_oa[tp];
    int c1;
    if (cand == HK) {
      c1 = oa;
    } else {
      int j = cand, neg = 0;
      if (j > HK) { j -= HK + 1; neg = 1; }
      const int i = j >> 3;
      const int r = (oa ^ (1 << i)) ^ (int)s_msk[j][tp];
      c1 = neg ? -r : r;
    }
    out[row * TP + tp]                 = (float)c1;   // selected_locs
    out[Bsz * T1 * TP + row * TP + tp] = mloss;       // min_loss
  }
}

extern "C" void kernel_launch(void* const* d_in, const int* in_sizes, int n_in,
                              void* d_out, int out_size, void* d_ws, size_t ws_size,
                              hipStream_t stream) {
  (void)in_sizes; (void)n_in; (void)out_size; (void)d_ws; (void)ws_size;
  const int*   sta       = (const int*)d_in[0];
  const int*   pos       = (const int*)d_in[1];
  const float* val_v     = (const float*)d_in[2];
  const float* val_n     = (const float*)d_in[3];
  const int*   locations = (const int*)d_in[4];
  const int*   epoch     = (const int*)d_in[5];
  float*       out       = (float*)d_out;

  dim3 grid(Bsz * T1);   // 256 workgroups, one per (b,t1)
  dim3 block(BLOCK);     // 8 wave32 waves
  hipLaunchKernelGGL(critigraph_kernel, grid, block, 0, stream,
                     sta, pos, val_v, val_n, locations, epoch, out);
}